// AvgPoolingInitializer_28707561407358
// MI455X (gfx1250) — compile-verified
//
#include <hip/hip_runtime.h>
#include <hip/hip_bf16.h>

typedef float v2f __attribute__((ext_vector_type(2)));
typedef float v8f __attribute__((ext_vector_type(8)));

// ---------------------------------------------------------------------------
// Stage 0: zero the reduction scratch (ssum, cnt, packed-argmax)
// ---------------------------------------------------------------------------
__global__ void zero_kernel(unsigned int* __restrict__ p, int n) {
    int t = blockIdx.x * 256 + threadIdx.x;
    if (t < n) p[t] = 0u;
}

// ---------------------------------------------------------------------------
// Stage 1: bilinear downsample (exact 2x2 average for scale 4/8/16 with
// half-pixel centers) -> threshold mask (f32 0/1) + per-(b,i) count and
// first-occurrence argmax of the soft mask.
// grid = (hw/256, B*I), block = 256
// ---------------------------------------------------------------------------
__global__ void resize_mask_kernel(const float* __restrict__ scr,      // [B,I,512,512]
                                   float* __restrict__ mask,           // [B,I,h,w]
                                   int* __restrict__ cnt,              // [B*I]
                                   unsigned long long* __restrict__ amax, // [B*I]
                                   int w, int s, int off) {
    const int bi = blockIdx.y;                       // b*16 + i
    const int k  = blockIdx.x * 256 + threadIdx.x;   // flat h*w index
    const int y  = k / w;
    const int x  = k - y * w;

    const float* S = scr + (size_t)bi * (512 * 512);
    const int r0 = y * s + off;
    const int c0 = x * s + off;
    const float* p = S + (size_t)r0 * 512 + c0;
    const float sr = 0.25f * (p[0] + p[1] + p[512] + p[513]);
    const float m  = (sr > 0.5f) ? 1.0f : 0.0f;

    mask[(size_t)bi * (blockDim.x * gridDim.x) + k] = m;

    // packed key: value in high bits (sr >= 0 so float bits are monotonic),
    // inverted index in low bits -> atomicMax picks max value, first index on tie
    unsigned long long key =
        ((unsigned long long)__float_as_uint(sr) << 32) |
        (unsigned long long)(0xFFFFFFFFu - (unsigned int)k);

    __shared__ int scnt[256];
    __shared__ unsigned long long skey[256];
    scnt[threadIdx.x] = (int)m;
    skey[threadIdx.x] = key;
    __syncthreads();
    for (int st = 128; st > 0; st >>= 1) {
        if ((int)threadIdx.x < st) {
            scnt[threadIdx.x] += scnt[threadIdx.x + st];
            if (skey[threadIdx.x + st] > skey[threadIdx.x])
                skey[threadIdx.x] = skey[threadIdx.x + st];
        }
        __syncthreads();
    }
    if (threadIdx.x == 0) {
        atomicAdd(&cnt[bi], scnt[0]);
        atomicMax(&amax[bi], skey[0]);
    }
}

// ---------------------------------------------------------------------------
// Stage 2: masked sum via V_WMMA_F32_16X16X4_F32.
//   ssum[b,i,c] += sum_k mask[b,i,k] * fmap[b,c,k]
// One wave (32 threads) per (batch, 64-channel group, K chunk).
// A (16x4 f32): lane&15 = row M; lanes>=16 hold K=2,3 (ksel).
// B (4x16 f32): lane&15 = col N (channel); lanes>=16 hold K=2,3.
// C/D (16x16 f32): 8 VGPRs, lanes>=16 hold rows M=8..15.
// KCHUNK is compile-time so the unrolled loop folds the 16B/step advance into
// load immediate offsets instead of per-step 64-bit address adds.
// ---------------------------------------------------------------------------
template <int KCHUNK>
__global__ void __launch_bounds__(32)
wmma_masked_sum_kernel(const float* __restrict__ fmap,   // [B,256,hw]
                       const float* __restrict__ mask,   // [B,16,hw]
                       float* __restrict__ ssum,         // [B,16,256]
                       int hw) {
    const int lane = threadIdx.x;
    const int l15  = lane & 15;
    const int half = lane >> 4;
    const int ksel = half << 1;                 // 0 or 2
    const int b    = blockIdx.x;                // batch
    const int cg   = blockIdx.y;                // 64-channel group
    const int k0   = blockIdx.z * KCHUNK;       // K chunk base

    const float* arow  = mask + ((size_t)(b * 16 + l15)) * hw + ksel + k0;
    const int cbase    = cg * 64 + l15;
    const float* brow0 = fmap + ((size_t)(b * 256 + cbase +  0)) * hw + ksel + k0;
    const float* brow1 = fmap + ((size_t)(b * 256 + cbase + 16)) * hw + ksel + k0;
    const float* brow2 = fmap + ((size_t)(b * 256 + cbase + 32)) * hw + ksel + k0;
    const float* brow3 = fmap + ((size_t)(b * 256 + cbase + 48)) * hw + ksel + k0;

    v8f acc0 = {}; v8f acc1 = {}; v8f acc2 = {}; v8f acc3 = {};
#pragma unroll 4
    for (int k = 0; k < KCHUNK; k += 4) {
        if ((k & 63) == 0) {
            // stream hint one stripe ahead (global_prefetch_b8); off the
            // critical dependency chain, once per 16 K-steps
            __builtin_prefetch((const void*)(brow0 + k + 64), 0, 1);
            __builtin_prefetch((const void*)(brow1 + k + 64), 0, 1);
            __builtin_prefetch((const void*)(brow2 + k + 64), 0, 1);
            __builtin_prefetch((const void*)(brow3 + k + 64), 0, 1);
        }
        v2f a  = *(const v2f*)(arow  + k);
        v2f b0 = *(const v2f*)(brow0 + k);
        v2f b1 = *(const v2f*)(brow1 + k);
        v2f b2 = *(const v2f*)(brow2 + k);
        v2f b3 = *(const v2f*)(brow3 + k);
        acc0 = __builtin_amdgcn_wmma_f32_16x16x4_f32(false, a, false, b0, (short)0, acc0, false, false);
        acc1 = __builtin_amdgcn_wmma_f32_16x16x4_f32(false, a, false, b1, (short)0, acc1, false, false);
        acc2 = __builtin_amdgcn_wmma_f32_16x16x4_f32(false, a, false, b2, (short)0, acc2, false, false);
        acc3 = __builtin_amdgcn_wmma_f32_16x16x4_f32(false, a, false, b3, (short)0, acc3, false, false);
    }

    float* obase = ssum + (size_t)(b * 16) * 256 + cg * 64 + l15;
#pragma unroll
    for (int v = 0; v < 8; ++v) {
        const int i = half * 8 + v;             // output row M
        float* o = obase + (size_t)i * 256;
        atomicAdd(o +  0, acc0[v]);
        atomicAdd(o + 16, acc1[v]);
        atomicAdd(o + 32, acc2[v]);
        atomicAdd(o + 48, acc3[v]);
    }
}

// ---------------------------------------------------------------------------
// Stage 3: per-element combine: mean = ssum / cnt (or argmax-gather fallback
// when cnt == 0), averaged over the 3 levels.
// ---------------------------------------------------------------------------
__global__ void combine_kernel(const float* __restrict__ f0, const float* __restrict__ f1,
                               const float* __restrict__ f2,
                               const float* __restrict__ s0, const float* __restrict__ s1,
                               const float* __restrict__ s2,
                               const int* __restrict__ cnt,
                               const unsigned long long* __restrict__ amax,
                               float* __restrict__ out) {
    const int t = blockIdx.x * 256 + threadIdx.x;
    if (t >= 8 * 16 * 256) return;
    const int c  = t & 255;
    const int bi = t >> 8;      // b*16 + i
    const int b  = bi >> 4;

    float r = 0.0f;
    {
        const int cl = cnt[bi];
        if (cl > 0) r += s0[t] / (float)cl;
        else {
            unsigned int k = 0xFFFFFFFFu - (unsigned int)(amax[bi] & 0xFFFFFFFFull);
            r += f0[((size_t)(b * 256 + c)) * 16384 + k];
        }
    }
    {
        const int cl = cnt[128 + bi];
        if (cl > 0) r += s1[t] / (float)cl;
        else {
            unsigned int k = 0xFFFFFFFFu - (unsigned int)(amax[128 + bi] & 0xFFFFFFFFull);
            r += f1[((size_t)(b * 256 + c)) * 4096 + k];
        }
    }
    {
        const int cl = cnt[256 + bi];
        if (cl > 0) r += s2[t] / (float)cl;
        else {
            unsigned int k = 0xFFFFFFFFu - (unsigned int)(amax[256 + bi] & 0xFFFFFFFFull);
            r += f2[((size_t)(b * 256 + c)) * 1024 + k];
        }
    }
    out[t] = r * (1.0f / 3.0f);
}

// ---------------------------------------------------------------------------
// Launcher
// inputs: feat0 [8,256,128,128] f32, feat1 [8,256,64,64] f32,
//         feat2 [8,256,32,32] f32, scribbles [8,16,512,512] f32
// output: [8,16,256] f32
// ---------------------------------------------------------------------------
extern "C" void kernel_launch(void* const* d_in, const int* in_sizes, int n_in,
                              void* d_out, int out_size, void* d_ws, size_t ws_size,
                              hipStream_t stream) {
    const float* feat0 = (const float*)d_in[0];
    const float* feat1 = (const float*)d_in[1];
    const float* feat2 = (const float*)d_in[2];
    const float* scr   = (const float*)d_in[3];
    float* out = (float*)d_out;

    // workspace layout (floats)
    float* mask0 = (float*)d_ws;                       // 8*16*16384
    float* mask1 = mask0 + (size_t)8 * 16 * 16384;     // 8*16*4096
    float* mask2 = mask1 + (size_t)8 * 16 * 4096;      // 8*16*1024
    float* ssum0 = mask2 + (size_t)8 * 16 * 1024;      // 8*16*256
    float* ssum1 = ssum0 + 8 * 16 * 256;
    float* ssum2 = ssum1 + 8 * 16 * 256;
    int*   cnt   = (int*)(ssum2 + 8 * 16 * 256);       // 3*128 ints
    unsigned long long* amax = (unsigned long long*)(cnt + 3 * 128); // 3*128 u64

    // zero ssum (3*32768 f32) + cnt (3*128 i32) + amax (3*128 u64)
    const int zwords = 3 * 32768 + 3 * 128 + 3 * 128 * 2;
    zero_kernel<<<(zwords + 255) / 256, 256, 0, stream>>>((unsigned int*)ssum0, zwords);

    // resize + threshold + count + argmax, per level
    resize_mask_kernel<<<dim3(16384 / 256, 128), 256, 0, stream>>>(scr, mask0, cnt +   0, amax +   0, 128,  4, 1);
    resize_mask_kernel<<<dim3( 4096 / 256, 128), 256, 0, stream>>>(scr, mask1, cnt + 128, amax + 128,  64,  8, 3);
    resize_mask_kernel<<<dim3( 1024 / 256, 128), 256, 0, stream>>>(scr, mask2, cnt + 256, amax + 256,  32, 16, 7);

    // masked sum via f32 WMMA; grid = (B, C/64, hw/KCHUNK), 1 wave per block
    wmma_masked_sum_kernel<512><<<dim3(8, 4, 32), 32, 0, stream>>>(feat0, mask0, ssum0, 16384);
    wmma_masked_sum_kernel<256><<<dim3(8, 4, 16), 32, 0, stream>>>(feat1, mask1, ssum1,  4096);
    wmma_masked_sum_kernel<128><<<dim3(8, 4,  8), 32, 0, stream>>>(feat2, mask2, ssum2,  1024);

    // combine: divide by counts (fallback gather when empty), average levels
    combine_kernel<<<(8 * 16 * 256 + 255) / 256, 256, 0, stream>>>(
        feat0, feat1, feat2, ssum0, ssum1, ssum2, cnt, amax, out);
}